// SCDLoss_37168646979884
// MI455X (gfx1250) — compile-verified
//
#include <hip/hip_runtime.h>

typedef float v4f __attribute__((ext_vector_type(4)));
typedef float v2f __attribute__((ext_vector_type(2)));
typedef float v8f __attribute__((ext_vector_type(8)));

// Hue of raw (un-normalized) RGB; hue is invariant under uniform scaling, so
// skipping the /255 matches the reference. Python `% 1.0` == h - floor(h).
__device__ __forceinline__ float hue_raw(float r, float g, float b) {
    float mx = fmaxf(r, fmaxf(g, b));
    float mn = fminf(r, fminf(g, b));
    float delta = mx - mn;
    float rd = __builtin_amdgcn_rcpf(delta);  // fast TRANS recip; delta==0 case masked below
    float h6 = (mx == r) ? (g - b) * rd
             : (mx == g) ? 2.0f + (b - r) * rd
                         : 4.0f + (r - g) * rd;
    float h = h6 * (1.0f / 6.0f);
    h = h - floorf(h);                         // python-style mod into [0,1)
    return (delta == 0.0f) ? 0.0f : h;
}

// Per-pixel hue loss term: diff if diff<0.5, diff-0.5 if diff>0.5, else 0.
__device__ __forceinline__ float hue_term(float hp, float ht) {
    float diff = fabsf(hp - ht);
    return (diff < 0.5f) ? diff : ((diff > 0.5f) ? diff - 0.5f : 0.0f);
}

// One (pred, target) pair, one quad of 4 pixels: 6 NT 128-bit loads.
__device__ __forceinline__ void accum_pair(const float* __restrict__ X,
                                           const float* __restrict__ G,
                                           size_t base, size_t HW,
                                           float& l1, float& hh) {
    v4f xr = __builtin_nontemporal_load((const v4f*)(X + base));
    v4f xg = __builtin_nontemporal_load((const v4f*)(X + base + HW));
    v4f xb = __builtin_nontemporal_load((const v4f*)(X + base + 2 * HW));
    v4f gr = __builtin_nontemporal_load((const v4f*)(G + base));
    v4f gg = __builtin_nontemporal_load((const v4f*)(G + base + HW));
    v4f gb = __builtin_nontemporal_load((const v4f*)(G + base + 2 * HW));
#pragma unroll
    for (int i = 0; i < 4; ++i) {
        l1 += fabsf(xr[i] - gr[i]) + fabsf(xg[i] - gg[i]) + fabsf(xb[i] - gb[i]);
        hh += hue_term(hue_raw(xr[i], xg[i], xb[i]), hue_raw(gr[i], gg[i], gb[i]));
    }
}

__global__ __launch_bounds__(256)
void scd_partial(const float* __restrict__ T, const float* __restrict__ R,
                 const float* __restrict__ GT, const float* __restrict__ GR,
                 float* __restrict__ blocksums,
                 unsigned int nq, unsigned int qpi, float s_l1, float s_hue) {
    unsigned int tid = blockIdx.x * blockDim.x + threadIdx.x;
    float l1 = 0.0f, hh = 0.0f;
    if (tid < nq) {
        unsigned int n  = tid / qpi;          // image index
        unsigned int qw = tid - n * qpi;      // quad index within channel plane
        size_t HW   = (size_t)qpi * 4;        // floats per channel plane
        size_t base = (size_t)n * 3 * HW + (size_t)qw * 4;
        accum_pair(T, GT, base, HW, l1, hh);  // (T, GT_T)
        accum_pair(R, GR, base, HW, l1, hh);  // (R, GT_R)
    }
    // Pre-scaled so one scalar per thread sums directly to the final answer.
    float p = l1 * s_l1 + hh * s_hue;

    // ---- wave32 reduction via V_WMMA_F32_16X16X4_F32 ----
    // With B = all-ones, C = 0:  D[m][n] = rowsum_m(A) for every n, so
    // sum(all 256 D entries) == 16 * sum(all A entries) for ANY lane->matrix
    // mapping. Put p in one A slot per lane, 0 in the other -> exact f32 sum.
    float t;
    float post_scale;
#if __has_builtin(__builtin_amdgcn_wmma_f32_16x16x4_f32)
    {
        v2f a;  a[0] = p;    a[1] = 0.0f;
        v2f bm; bm[0] = 1.0f; bm[1] = 1.0f;
        v8f c = {};
        v8f d = __builtin_amdgcn_wmma_f32_16x16x4_f32(
            /*neg_a=*/false, a, /*neg_b=*/false, bm,
            /*c_mod=*/(short)0, c, /*reuse_a=*/false, /*reuse_b=*/false);
        t = ((d[0] + d[1]) + (d[2] + d[3])) + ((d[4] + d[5]) + (d[6] + d[7]));
        post_scale = 1.0f / 16.0f;            // undo the 16x replication across N
    }
#else
    t = p;
    post_scale = 1.0f;
#endif
    // fold the per-lane values across the 32 lanes (fixed order -> deterministic)
    t += __shfl_xor(t, 16, 32);
    t += __shfl_xor(t, 8, 32);
    t += __shfl_xor(t, 4, 32);
    t += __shfl_xor(t, 2, 32);
    t += __shfl_xor(t, 1, 32);

    __shared__ float red[8];                  // 256 threads = 8 wave32
    unsigned int lane = threadIdx.x & 31u;
    unsigned int wv   = threadIdx.x >> 5;
    if (lane == 0) red[wv] = t * post_scale;
    __syncthreads();
    if (threadIdx.x == 0) {
        float s = 0.0f;
#pragma unroll
        for (int i = 0; i < 8; ++i) s += red[i];
        blocksums[blockIdx.x] = s;
    }
}

__global__ __launch_bounds__(256)
void scd_final(const float* __restrict__ blocksums, int nb, float* __restrict__ out) {
    __shared__ float sm[256];
    int tid = threadIdx.x;
    float s = 0.0f;
    for (int i = tid; i < nb; i += 256) s += blocksums[i];  // fixed strided order
    sm[tid] = s;
    __syncthreads();
#pragma unroll
    for (int off = 128; off > 0; off >>= 1) {
        if (tid < off) sm[tid] += sm[tid + off];
        __syncthreads();
    }
    if (tid == 0) out[0] = sm[0];
}

extern "C" void kernel_launch(void* const* d_in, const int* in_sizes, int n_in,
                              void* d_out, int out_size, void* d_ws, size_t ws_size,
                              hipStream_t stream) {
    // setup_inputs() order: T, R, GT_T, GT_R  (all float32)
    const float* T  = (const float*)d_in[0];
    const float* R  = (const float*)d_in[1];
    const float* GT = (const float*)d_in[2];
    const float* GR = (const float*)d_in[3];
    float* out       = (float*)d_out;
    float* blocksums = (float*)d_ws;

    const long HW   = 512L * 512L;            // per reference shape (B,3,512,512)
    long E    = in_sizes[0];                  // B*3*HW elements
    long npix = E / 3;                        // B*HW pixels
    long nq   = npix / 4;                     // quads (4 pixels each)
    unsigned int qpi = (unsigned int)(HW / 4);

    int threads = 256;
    int blocks  = (int)((nq + threads - 1) / threads);  // 4096 for the ref shape

    // out = 0.25*(l1_T+l1_R+hsv_T+hsv_R)
    //     = sum_raw_l1 * 0.25/(255*3*npix) + sum_raw_hue * 0.25/npix
    float s_l1  = (float)(0.25 / (765.0 * (double)npix));
    float s_hue = (float)(0.25 / (double)npix);

    scd_partial<<<blocks, threads, 0, stream>>>(T, R, GT, GR, blocksums,
                                                (unsigned int)nq, qpi, s_l1, s_hue);
    scd_final<<<1, 256, 0, stream>>>(blocksums, blocks, out);
}